// EOSFocusedLoss_9483287789848
// MI455X (gfx1250) — compile-verified
//
#include <hip/hip_runtime.h>
#include <math.h>

typedef __attribute__((ext_vector_type(2))) float v2f;
typedef __attribute__((ext_vector_type(8))) float v8f;

#define NB   32      // batch
#define NS   256     // sequence
#define NV   16000   // vocab
#define NROWS (NB * NS)

// --------------------------------------------------------------------------
// Kernel 1: fused per-row online-softmax + argmax.
// One block per (b,s) row. Streams 16000 f32 as float4, maintains running
// max m, rescaled sum s (1 exp/element amortized), and argmax index.
// Block reduce: wave32 shfl max+idx, then WMMA f32 16x16x4 (ones-B) to sum
// the 256 rescaled partials exactly in f32.
// --------------------------------------------------------------------------
__global__ __launch_bounds__(256) void k_row_softmax(
    const float* __restrict__ logits,
    const long long* __restrict__ tgt,
    float* __restrict__ nll_out,
    int* __restrict__ pred_out)
{
  const int row = blockIdx.x;
  const int tid = threadIdx.x;
  const float* rowf = logits + (size_t)row * NV;
  const float4* rowp = reinterpret_cast<const float4*>(rowf);

  float m = -INFINITY;
  float s = 0.0f;
  int   idx = 0;

  for (int j = tid; j < NV / 4; j += 256) {
    float4 v = rowp[j];
    __builtin_prefetch(rowp + j + 512, 0, 0);   // gfx1250 global_prefetch_b8
    const int base = j * 4;
    {
      float x = v.x;
      if (x > m) { s = s * __expf(m - x) + 1.0f; m = x; idx = base + 0; }
      else       { s += __expf(x - m); }
    }
    {
      float x = v.y;
      if (x > m) { s = s * __expf(m - x) + 1.0f; m = x; idx = base + 1; }
      else       { s += __expf(x - m); }
    }
    {
      float x = v.z;
      if (x > m) { s = s * __expf(m - x) + 1.0f; m = x; idx = base + 2; }
      else       { s += __expf(x - m); }
    }
    {
      float x = v.w;
      if (x > m) { s = s * __expf(m - x) + 1.0f; m = x; idx = base + 3; }
      else       { s += __expf(x - m); }
    }
  }

  const float lm = m;  // keep thread-local max for rescale

  // ---- wave32 max+index reduction (first-occurrence tie break) ----
  float rm = m; int ri = idx;
  #pragma unroll
  for (int off = 16; off > 0; off >>= 1) {
    float om = __shfl_xor(rm, off, 32);
    int   oi = __shfl_xor(ri, off, 32);
    if (om > rm || (om == rm && oi < ri)) { rm = om; ri = oi; }
  }

  __shared__ float swm[8];
  __shared__ int   swi[8];
  __shared__ float s_bm;
  __shared__ int   s_bi;
  __shared__ float vals[256];

  const int wave = tid >> 5;
  if ((tid & 31) == 0) { swm[wave] = rm; swi[wave] = ri; }
  __syncthreads();
  if (tid == 0) {
    float bm = swm[0]; int bi = swi[0];
    #pragma unroll
    for (int w = 1; w < 8; ++w) {
      if (swm[w] > bm || (swm[w] == bm && swi[w] < bi)) { bm = swm[w]; bi = swi[w]; }
    }
    s_bm = bm; s_bi = bi;
  }
  __syncthreads();
  const float bm = s_bm;

  // rescale each thread's partial sum to the block max; lay out as 16x16
  vals[tid] = s * __expf(lm - bm);
  __syncthreads();

  // ---- WMMA reduction of 256 f32 partials: 4x v_wmma_f32_16x16x4_f32 ----
  // A[m][k] = vals[m*16 + 4*chunk + k], B = ones(4x16) -> D[m][n] = rowsum[m]
  if (tid < 32) {
    const int lane = tid;
    const int mrow = lane & 15;
    const int kb   = (lane < 16) ? 0 : 2;   // A-matrix 16x4 f32 lane layout
    v8f c = {0.0f, 0.0f, 0.0f, 0.0f, 0.0f, 0.0f, 0.0f, 0.0f};
    v2f ones = {1.0f, 1.0f};
    #pragma unroll
    for (int ch = 0; ch < 4; ++ch) {
      v2f a;
      a[0] = vals[mrow * 16 + ch * 4 + kb + 0];
      a[1] = vals[mrow * 16 + ch * 4 + kb + 1];
      c = __builtin_amdgcn_wmma_f32_16x16x4_f32(
          /*neg_a=*/false, a, /*neg_b=*/false, ones,
          /*c_mod=*/(short)0, c, /*reuse_a=*/false, /*reuse_b=*/false);
    }
    // lanes 0-15 hold rowsum[0..7] in c[0..7]; lanes 16-31 hold rowsum[8..15]
    float part = c[0] + c[1] + c[2] + c[3] + c[4] + c[5] + c[6] + c[7];
    float total = part + __shfl_xor(part, 16, 32);

    if (tid == 0) {
      long long t = tgt[row];
      float xt = rowf[(int)t];
      nll_out[row]  = -(xt - bm - __logf(total));
      pred_out[row] = s_bi;
    }
  }
}

// --------------------------------------------------------------------------
// Kernel 2: per-batch stats. One block per batch row.
// --------------------------------------------------------------------------
__global__ __launch_bounds__(256) void k_batch_stats(
    const long long* __restrict__ tgt,
    const float* __restrict__ nll,
    const int* __restrict__ preds,
    float* __restrict__ b_nllsum,
    float* __restrict__ b_keep,
    float* __restrict__ b_predlen,
    float* __restrict__ b_rep,
    float* __restrict__ b_haseos,
    int* __restrict__ b_eospos)
{
  const int b = blockIdx.x;
  const int tid = threadIdx.x;

  __shared__ int   pr[NS];
  __shared__ int   sh_L, sh_E, sh_rep;
  __shared__ float red[NS];

  const int p = preds[b * NS + tid];
  const long long tv = tgt[b * NS + tid];
  pr[tid] = p;
  if (tid == 0) { sh_L = NS; sh_E = NS; sh_rep = 0; }
  __syncthreads();

  if (p == 0)  atomicMin(&sh_L, tid);   // first PAD in predictions
  if (tv == 1) atomicMin(&sh_E, tid);   // first EOS in targets
  __syncthreads();
  const int L = sh_L;

  // repetition windows for periods 2,3,4 (each thread = one start index)
  bool myrep = false;
  #pragma unroll
  for (int per = 2; per <= 4; ++per) {
    if (tid <= NS - 3 * per) {
      bool full = true;
      for (int i = 0; i < 2 * per; ++i)
        full = full && (pr[tid + i] == pr[tid + i + per]);
      if (full && (tid + 3 * per <= L) && (L >= 3 * per + 3)) myrep = true;
    }
  }
  if (myrep) sh_rep = 1;

  const float keep = (tv != 0) ? 1.0f : 0.0f;
  const float vnll = nll[b * NS + tid] * keep;
  const float plen = (p != 0) ? 1.0f : 0.0f;

  // three sequential block sums
  red[tid] = vnll; __syncthreads();
  for (int o = 128; o > 0; o >>= 1) { if (tid < o) red[tid] += red[tid + o]; __syncthreads(); }
  const float nlls = red[0]; __syncthreads();

  red[tid] = keep; __syncthreads();
  for (int o = 128; o > 0; o >>= 1) { if (tid < o) red[tid] += red[tid + o]; __syncthreads(); }
  const float keeps = red[0]; __syncthreads();

  red[tid] = plen; __syncthreads();
  for (int o = 128; o > 0; o >>= 1) { if (tid < o) red[tid] += red[tid + o]; __syncthreads(); }
  const float plens = red[0];

  if (tid == 0) {
    b_nllsum[b]  = nlls;
    b_keep[b]    = keeps;
    b_predlen[b] = plens;
    b_rep[b]     = (float)sh_rep;
    b_haseos[b]  = (sh_E < NS) ? 1.0f : 0.0f;
    b_eospos[b]  = (sh_E < NS) ? sh_E : 0;
  }
}

// --------------------------------------------------------------------------
// Kernel 3: EOS margin per batch (rescan of one row per batch: ~2MB total).
// --------------------------------------------------------------------------
__global__ __launch_bounds__(256) void k_eos(
    const float* __restrict__ logits,
    const int* __restrict__ preds,
    const int* __restrict__ b_eospos,
    const float* __restrict__ b_haseos,
    float* __restrict__ b_margin,
    float* __restrict__ b_eospred)
{
  const int b = blockIdx.x;
  const int tid = threadIdx.x;
  const int pos = b_eospos[b];
  const float* row = logits + ((size_t)b * NS + pos) * NV;

  float mo = -INFINITY;
  for (int i = tid; i < NV; i += 256) {
    float x = row[i];
    if (i != 1) mo = fmaxf(mo, x);   // max over non-EOS vocab entries
  }

  __shared__ float red[256];
  red[tid] = mo; __syncthreads();
  for (int o = 128; o > 0; o >>= 1) {
    if (tid < o) red[tid] = fmaxf(red[tid], red[tid + o]);
    __syncthreads();
  }

  if (tid == 0) {
    const float has  = b_haseos[b];
    const float eosl = row[1];
    const float margin = fmaxf(red[0] - eosl + 1.0f, 0.0f);
    b_margin[b]  = (has != 0.0f) ? margin : 0.0f;
    const int pa = preds[b * NS + pos];
    b_eospred[b] = (has != 0.0f && pa == 1) ? 1.0f : 0.0f;
  }
}

// --------------------------------------------------------------------------
// Kernel 4: final wave32 reduction over the 32 batches -> 6 scalars.
// --------------------------------------------------------------------------
__global__ __launch_bounds__(32) void k_final(
    const float* __restrict__ b_nllsum,
    const float* __restrict__ b_keep,
    const float* __restrict__ b_predlen,
    const float* __restrict__ b_rep,
    const float* __restrict__ b_haseos,
    const float* __restrict__ b_margin,
    const float* __restrict__ b_eospred,
    float* __restrict__ out)
{
  const int lane = threadIdx.x;
  float nlls = b_nllsum[lane];
  float keep = b_keep[lane];
  float plen = b_predlen[lane];
  float rep  = b_rep[lane];
  float has  = b_haseos[lane];
  float marg = b_margin[lane];
  float ep   = b_eospred[lane];

  #pragma unroll
  for (int o = 16; o > 0; o >>= 1) {
    nlls += __shfl_xor(nlls, o, 32);
    keep += __shfl_xor(keep, o, 32);
    plen += __shfl_xor(plen, o, 32);
    rep  += __shfl_xor(rep,  o, 32);
    has  += __shfl_xor(has,  o, 32);
    marg += __shfl_xor(marg, o, 32);
    ep   += __shfl_xor(ep,   o, 32);
  }

  if (lane == 0) {
    const float main_loss = nlls / fmaxf(keep, 1.0f);
    const float pattern   = rep / (float)NB * 100.0f;
    const float eos_loss  = marg / (float)NB;
    const float rate      = ep / fmaxf(has, 1.0f);
    const float avg_p     = plen / (float)NB;
    const float avg_t     = keep / (float)NB;
    const float lp        = fabsf(avg_p - avg_t) / avg_t;
    const float total     = main_loss + 20.0f * eos_loss + 2.0f * pattern + 0.5f * lp;
    out[0] = total;
    out[1] = main_loss;
    out[2] = eos_loss;
    out[3] = pattern;
    out[4] = lp;
    out[5] = rate;
  }
}

// --------------------------------------------------------------------------
extern "C" void kernel_launch(void* const* d_in, const int* in_sizes, int n_in,
                              void* d_out, int out_size, void* d_ws, size_t ws_size,
                              hipStream_t stream) {
  (void)in_sizes; (void)n_in; (void)out_size; (void)ws_size;
  const float* logits   = (const float*)d_in[0];
  const long long* tgt  = (const long long*)d_in[1];   // int64 targets
  float* out = (float*)d_out;

  char* ws = (char*)d_ws;
  float* nll       = (float*)(ws);                 // 8192 f32
  int*   preds     = (int*)  (ws + 32768);         // 8192 i32
  float* b_nllsum  = (float*)(ws + 65536 + 0);
  float* b_keep    = (float*)(ws + 65536 + 128);
  float* b_predlen = (float*)(ws + 65536 + 256);
  float* b_rep     = (float*)(ws + 65536 + 384);
  float* b_haseos  = (float*)(ws + 65536 + 512);
  float* b_margin  = (float*)(ws + 65536 + 640);
  float* b_eospred = (float*)(ws + 65536 + 768);
  int*   b_eospos  = (int*)  (ws + 65536 + 896);

  k_row_softmax<<<NROWS, 256, 0, stream>>>(logits, tgt, nll, preds);
  k_batch_stats<<<NB, 256, 0, stream>>>(tgt, nll, preds, b_nllsum, b_keep,
                                        b_predlen, b_rep, b_haseos, b_eospos);
  k_eos<<<NB, 256, 0, stream>>>(logits, preds, b_eospos, b_haseos,
                                b_margin, b_eospred);
  k_final<<<1, 32, 0, stream>>>(b_nllsum, b_keep, b_predlen, b_rep,
                                b_haseos, b_margin, b_eospred, out);
}